// cWCT_36739150250149
// MI455X (gfx1250) — compile-verified
//
#include <hip/hip_runtime.h>

// Problem constants from the reference: B=8, N=512, H*W=4096, fp32.
#define BB 8
#define NN 512
#define MM 4096
#define LDP 66    // LDS K-pitch (floats): 64 + 2 pad to break 64-bank conflicts
#define LDPN 130  // LDS P-pitch for the k-major B panel in the NN kernel

typedef float v2f __attribute__((ext_vector_type(2)));
typedef float v8f __attribute__((ext_vector_type(8)));

// ---------------------------------------------------------------------------
// Wave-level WMMA K-loops. 256-thread workgroup = 8 waves arranged 2x4 over a
// 64(row) x 128(col) macro tile. Each wave owns a 32x32 tile = 2x2 WMMA tiles
// with 4 INDEPENDENT f32 accumulators (breaks the WMMA->WMMA RAW chain) and
// 2x fragment reuse (halves LDS reads per WMMA).
// A fragment (16x4 f32): lane L -> row L%16, K = 2*(L/16) + {0,1}
// C/D (16x16 f32): VGPR r -> row r + 8*(L>=16), col = L%16
// ---------------------------------------------------------------------------

// acc[ti][tj] += sum_k A[i,k] * Bt[j,k]  (both row-major, K contiguous)
__device__ __forceinline__ void wmma_kloop_nt(const float* __restrict__ Arows,
                                              const float* __restrict__ Brows,
                                              int lda, int ldb, int K,
                                              float (&lA)[64][LDP],
                                              float (&lB)[128][LDP],
                                              v8f (&acc)[2][2])
{
  const int tid  = threadIdx.x;
  const int lane = tid & 31;
  const int wid  = tid >> 5;
  const int wr   = wid >> 2, wc = wid & 3;
  const int ar   = wr * 32 + (lane & 15);
  const int br   = wc * 32 + (lane & 15);
  const int kh   = (lane >> 4) * 2;
  for (int k0 = 0; k0 < K; k0 += 64) {
    // Stage A panel: 64 rows x 64 K (coalesced float4 loads, 4 per thread)
    for (int q = tid; q < 1024; q += 256) {
      int row = q >> 4, c4 = (q & 15) << 2;
      float4 v = *(const float4*)(Arows + (size_t)row * lda + k0 + c4);
      lA[row][c4] = v.x; lA[row][c4 + 1] = v.y; lA[row][c4 + 2] = v.z; lA[row][c4 + 3] = v.w;
    }
    // Stage B panel: 128 rows x 64 K (8 per thread)
    for (int q = tid; q < 2048; q += 256) {
      int row = q >> 4, c4 = (q & 15) << 2;
      float4 v = *(const float4*)(Brows + (size_t)row * ldb + k0 + c4);
      lB[row][c4] = v.x; lB[row][c4 + 1] = v.y; lB[row][c4 + 2] = v.z; lB[row][c4 + 3] = v.w;
    }
    // Prefetch next K panels into cache while this one is consumed
    if (k0 + 64 < K) {
      __builtin_prefetch(Arows + (size_t)(tid & 63) * lda + k0 + 64, 0, 3);
      __builtin_prefetch(Brows + (size_t)(tid & 127) * ldb + k0 + 64, 0, 3);
    }
    __syncthreads();
#pragma unroll
    for (int kk = 0; kk < 64; kk += 4) {
      v2f a0, a1, b0, b1;
      a0.x = lA[ar][kk + kh];      a0.y = lA[ar][kk + kh + 1];
      a1.x = lA[ar + 16][kk + kh]; a1.y = lA[ar + 16][kk + kh + 1];
      b0.x = lB[br][kk + kh];      b0.y = lB[br][kk + kh + 1];
      b1.x = lB[br + 16][kk + kh]; b1.y = lB[br + 16][kk + kh + 1];
      acc[0][0] = __builtin_amdgcn_wmma_f32_16x16x4_f32(false, a0, false, b0, (short)0, acc[0][0], false, false);
      acc[0][1] = __builtin_amdgcn_wmma_f32_16x16x4_f32(false, a0, false, b1, (short)0, acc[0][1], false, false);
      acc[1][0] = __builtin_amdgcn_wmma_f32_16x16x4_f32(false, a1, false, b0, (short)0, acc[1][0], false, false);
      acc[1][1] = __builtin_amdgcn_wmma_f32_16x16x4_f32(false, a1, false, b1, (short)0, acc[1][1], false, false);
    }
    __syncthreads();
  }
}

// acc[ti][tj] += sum_k A[i,k] * B[k,p]  (B panel staged k-major in LDS)
__device__ __forceinline__ void wmma_kloop_nn(const float* __restrict__ Arows,
                                              const float* __restrict__ Bcols,
                                              int lda, int ldb, int K,
                                              float (&lA)[64][LDP],
                                              float (&lB)[64][LDPN],
                                              v8f (&acc)[2][2])
{
  const int tid  = threadIdx.x;
  const int lane = tid & 31;
  const int wid  = tid >> 5;
  const int wr   = wid >> 2, wc = wid & 3;
  const int ar   = wr * 32 + (lane & 15);
  const int pc   = wc * 32 + (lane & 15);
  const int kh   = (lane >> 4) * 2;
  for (int k0 = 0; k0 < K; k0 += 64) {
    for (int q = tid; q < 1024; q += 256) {
      int row = q >> 4, c4 = (q & 15) << 2;
      float4 v = *(const float4*)(Arows + (size_t)row * lda + k0 + c4);
      lA[row][c4] = v.x; lA[row][c4 + 1] = v.y; lA[row][c4 + 2] = v.z; lA[row][c4 + 3] = v.w;
    }
    for (int q = tid; q < 2048; q += 256) {
      int row = q >> 5, c4 = (q & 31) << 2;   // row indexes K, 128 P-cols per row
      float4 v = *(const float4*)(Bcols + (size_t)(k0 + row) * ldb + c4);
      lB[row][c4] = v.x; lB[row][c4 + 1] = v.y; lB[row][c4 + 2] = v.z; lB[row][c4 + 3] = v.w;
    }
    if (k0 + 64 < K) {
      __builtin_prefetch(Arows + (size_t)(tid & 63) * lda + k0 + 64, 0, 3);
      __builtin_prefetch(Bcols + (size_t)(k0 + 64 + (tid & 63)) * ldb + (tid >> 6) * 64, 0, 3);
    }
    __syncthreads();
#pragma unroll
    for (int kk = 0; kk < 64; kk += 4) {
      v2f a0, a1, b0, b1;
      a0.x = lA[ar][kk + kh];          a0.y = lA[ar][kk + kh + 1];
      a1.x = lA[ar + 16][kk + kh];     a1.y = lA[ar + 16][kk + kh + 1];
      b0.x = lB[kk + kh][pc];          b0.y = lB[kk + kh + 1][pc];
      b1.x = lB[kk + kh][pc + 16];     b1.y = lB[kk + kh + 1][pc + 16];
      acc[0][0] = __builtin_amdgcn_wmma_f32_16x16x4_f32(false, a0, false, b0, (short)0, acc[0][0], false, false);
      acc[0][1] = __builtin_amdgcn_wmma_f32_16x16x4_f32(false, a0, false, b1, (short)0, acc[0][1], false, false);
      acc[1][0] = __builtin_amdgcn_wmma_f32_16x16x4_f32(false, a1, false, b0, (short)0, acc[1][0], false, false);
      acc[1][1] = __builtin_amdgcn_wmma_f32_16x16x4_f32(false, a1, false, b1, (short)0, acc[1][1], false, false);
    }
    __syncthreads();
  }
}

// ---------------------------------------------------------------------------
// 1) Row means: mc[b,n] = mean(cont[b,n,:]), ms likewise. grid(512, 8, 2)x256
// ---------------------------------------------------------------------------
__global__ void mean_kernel(const float* __restrict__ cont, const float* __restrict__ styl,
                            float* __restrict__ mc, float* __restrict__ ms)
{
  const int n = blockIdx.x, b = blockIdx.y, which = blockIdx.z;
  const float* X = (which ? styl : cont) + ((size_t)b * NN + n) * MM;
  float s = 0.f;
  for (int i = threadIdx.x; i < MM; i += 256) s += X[i];
  __shared__ float red[256];
  red[threadIdx.x] = s;
  __syncthreads();
  for (int off = 128; off > 0; off >>= 1) {
    if (threadIdx.x < off) red[threadIdx.x] += red[threadIdx.x + off];
    __syncthreads();
  }
  if (threadIdx.x == 0) (which ? ms : mc)[b * NN + n] = red[0] * (1.0f / MM);
}

// ---------------------------------------------------------------------------
// 2) Covariance: cov = (X X^T - M mu mu^T) / (M-1). grid(4, 8, 16)x256
// ---------------------------------------------------------------------------
__global__ void cov_kernel(const float* __restrict__ cont, const float* __restrict__ styl,
                           const float* __restrict__ mc, const float* __restrict__ ms,
                           float* __restrict__ covc, float* __restrict__ covs)
{
  __shared__ float lA[64][LDP];
  __shared__ float lB[128][LDP];
  const int zb = blockIdx.z;
  const int b = zb & 7, which = zb >> 3;
  const float* X  = (which ? styl : cont) + (size_t)b * NN * MM;
  const float* mu = (which ? ms : mc) + b * NN;
  float* C        = (which ? covs : covc) + (size_t)b * NN * NN;
  const int j0 = blockIdx.x * 128;
  const int i0 = blockIdx.y * 64;

  v8f acc[2][2] = {};
  wmma_kloop_nt(X + (size_t)i0 * MM, X + (size_t)j0 * MM, MM, MM, MM, lA, lB, acc);

  const int lane = threadIdx.x & 31, wid = threadIdx.x >> 5;
  const int wr = wid >> 2, wc = wid & 3;
  const float inv_m1 = 1.0f / (float)(MM - 1);
#pragma unroll
  for (int ti = 0; ti < 2; ++ti) {
    const int ibase = i0 + wr * 32 + ti * 16 + (lane >> 4) * 8;
#pragma unroll
    for (int tj = 0; tj < 2; ++tj) {
      const int j = j0 + wc * 32 + tj * 16 + (lane & 15);
      const float muj = mu[j];
#pragma unroll
      for (int r = 0; r < 8; ++r) {
        const int i = ibase + r;
        C[(size_t)i * NN + j] = (acc[ti][tj][r] - (float)MM * mu[i] * muj) * inv_m1;
      }
    }
  }
}

// ---------------------------------------------------------------------------
// 3) In-place Cholesky (right-looking, lower), then zero strict upper.
//    grid(8, 2) x 512; one workgroup per matrix (O(N^3/3), tiny vs GEMMs).
// ---------------------------------------------------------------------------
__global__ void chol_kernel(float* __restrict__ covc, float* __restrict__ covs)
{
  const int b = blockIdx.x, which = blockIdx.y;
  float* A = (which ? covs : covc) + (size_t)b * NN * NN;
  __shared__ float col[NN];
  __shared__ float dshared;
  const int t = threadIdx.x, T = blockDim.x;
  const int wid = t >> 5, lane = t & 31, NW = T >> 5;
  for (int j = 0; j < NN; ++j) {
    if (t == 0) {
      float d = sqrtf(A[(size_t)j * NN + j]);
      A[(size_t)j * NN + j] = d;
      dshared = 1.0f / d;
    }
    __syncthreads();
    const float dinv = dshared;
    for (int i = j + 1 + t; i < NN; i += T) {
      float v = A[(size_t)i * NN + j] * dinv;
      A[(size_t)i * NN + j] = v;
      col[i] = v;
    }
    __syncthreads();
    for (int i = j + 1 + wid; i < NN; i += NW) {
      const float ci = col[i];
      const size_t ro = (size_t)i * NN;
      for (int k = j + 1 + lane; k <= i; k += 32) A[ro + k] -= ci * col[k];
    }
    __syncthreads();
  }
  for (int idx = t; idx < NN * NN; idx += T) {
    int i = idx >> 9, jj = idx & (NN - 1);
    if (jj > i) A[idx] = 0.f;
  }
}

// ---------------------------------------------------------------------------
// 4) Triangular inverse, stored TRANSPOSED: XT[j,k] = invL[k,j]. grid(8) x 512
// ---------------------------------------------------------------------------
__global__ void trinv_kernel(const float* __restrict__ L, float* __restrict__ XT)
{
  const int b = blockIdx.x;
  const float* Lb = L + (size_t)b * NN * NN;
  float* X = XT + (size_t)b * NN * NN;
  const int t = threadIdx.x, T = blockDim.x;
  const int wid = t >> 5, lane = t & 31, NW = T >> 5;
  for (int idx = t; idx < NN * NN; idx += T) X[idx] = 0.f;
  __syncthreads();
  __shared__ float Lrow[NN];
  for (int i = 0; i < NN; ++i) {
    for (int k = t; k <= i; k += T) Lrow[k] = Lb[(size_t)i * NN + k];
    __syncthreads();
    const float dinv = 1.0f / Lrow[i];
    for (int j = wid; j <= i; j += NW) {
      float s = 0.f;
      for (int k = j + lane; k < i; k += 32) s += Lrow[k] * X[(size_t)j * NN + k];
#pragma unroll
      for (int off = 16; off > 0; off >>= 1) s += __shfl_down(s, off, 32);
      if (lane == 0) X[(size_t)j * NN + i] = (j == i) ? dinv : -s * dinv;
    }
    __syncthreads();
  }
}

// ---------------------------------------------------------------------------
// 5) T = Ls @ invLc  ==  nt-GEMM with Bt = invLc^T. grid(4, 8, 8)x256
// ---------------------------------------------------------------------------
__global__ void tmat_kernel(const float* __restrict__ Ls, const float* __restrict__ invLT,
                            float* __restrict__ Tm)
{
  __shared__ float lA[64][LDP];
  __shared__ float lB[128][LDP];
  const int b = blockIdx.z;
  const float* Ab = Ls + (size_t)b * NN * NN;
  const float* Bb = invLT + (size_t)b * NN * NN;
  float* Cb = Tm + (size_t)b * NN * NN;
  const int j0 = blockIdx.x * 128;
  const int i0 = blockIdx.y * 64;

  v8f acc[2][2] = {};
  wmma_kloop_nt(Ab + (size_t)i0 * NN, Bb + (size_t)j0 * NN, NN, NN, NN, lA, lB, acc);

  const int lane = threadIdx.x & 31, wid = threadIdx.x >> 5;
  const int wr = wid >> 2, wc = wid & 3;
#pragma unroll
  for (int ti = 0; ti < 2; ++ti) {
    const int ibase = i0 + wr * 32 + ti * 16 + (lane >> 4) * 8;
#pragma unroll
    for (int tj = 0; tj < 2; ++tj) {
      const int j = j0 + wc * 32 + tj * 16 + (lane & 15);
#pragma unroll
      for (int r = 0; r < 8; ++r) Cb[(size_t)(ibase + r) * NN + j] = acc[ti][tj][r];
    }
  }
}

// ---------------------------------------------------------------------------
// 6) bias[b,n] = ms[b,n] - dot(T[b,n,:], mc[b,:]). grid(8) x 512
// ---------------------------------------------------------------------------
__global__ void bias_kernel(const float* __restrict__ Tm, const float* __restrict__ mc,
                            const float* __restrict__ ms, float* __restrict__ bias)
{
  const int b = blockIdx.x, t = threadIdx.x;
  __shared__ float mu[NN];
  for (int k = t; k < NN; k += blockDim.x) mu[k] = mc[b * NN + k];
  __syncthreads();
  for (int n = t; n < NN; n += blockDim.x) {
    const float* Trow = Tm + ((size_t)b * NN + n) * NN;
    float s = 0.f;
    for (int k = 0; k < NN; ++k) s += Trow[k] * mu[k];
    bias[b * NN + n] = ms[b * NN + n] - s;
  }
}

// ---------------------------------------------------------------------------
// 7) out = T @ cont + bias (per-row). grid(32, 8, 8)x256
// ---------------------------------------------------------------------------
__global__ void final_kernel(const float* __restrict__ Tm, const float* __restrict__ cont,
                             const float* __restrict__ bias, float* __restrict__ out)
{
  __shared__ float lA[64][LDP];
  __shared__ float lB[64][LDPN];
  const int b = blockIdx.z;
  const float* Ab = Tm + (size_t)b * NN * NN;
  const float* Bb = cont + (size_t)b * NN * MM;   // [K=N rows][M cols]
  float* Cb = out + (size_t)b * NN * MM;
  const int p0 = blockIdx.x * 128;
  const int n0 = blockIdx.y * 64;

  v8f acc[2][2] = {};
  wmma_kloop_nn(Ab + (size_t)n0 * NN, Bb + p0, NN, MM, NN, lA, lB, acc);

  const int lane = threadIdx.x & 31, wid = threadIdx.x >> 5;
  const int wr = wid >> 2, wc = wid & 3;
#pragma unroll
  for (int ti = 0; ti < 2; ++ti) {
    const int nbase = n0 + wr * 32 + ti * 16 + (lane >> 4) * 8;
#pragma unroll
    for (int tj = 0; tj < 2; ++tj) {
      const int p = p0 + wc * 32 + tj * 16 + (lane & 15);
#pragma unroll
      for (int r = 0; r < 8; ++r) {
        const int n = nbase + r;
        Cb[(size_t)n * MM + p] = acc[ti][tj][r] + bias[b * NN + n];
      }
    }
  }
}

// ---------------------------------------------------------------------------
// Launcher. Workspace (floats): covc | covs | invLT | T (8*512*512 each)
// | mc | ms | bias (8*512 each)  (~33.6 MB total)
// ---------------------------------------------------------------------------
extern "C" void kernel_launch(void* const* d_in, const int* in_sizes, int n_in,
                              void* d_out, int out_size, void* d_ws, size_t ws_size,
                              hipStream_t stream)
{
  const float* cont = (const float*)d_in[0];
  const float* styl = (const float*)d_in[1];
  // d_in[2] (x_real) is reshaped but unused by the reference computation.
  float* out = (float*)d_out;

  float* covc  = (float*)d_ws;
  float* covs  = covc  + (size_t)BB * NN * NN;
  float* invLT = covs  + (size_t)BB * NN * NN;
  float* Tm    = invLT + (size_t)BB * NN * NN;
  float* mc    = Tm    + (size_t)BB * NN * NN;
  float* ms    = mc    + BB * NN;
  float* bias  = ms    + BB * NN;

  mean_kernel <<<dim3(NN, BB, 2), 256, 0, stream>>>(cont, styl, mc, ms);
  cov_kernel  <<<dim3(NN / 128, NN / 64, 2 * BB), 256, 0, stream>>>(cont, styl, mc, ms,
                                                                    covc, covs);
  chol_kernel <<<dim3(BB, 2), 512, 0, stream>>>(covc, covs);
  trinv_kernel<<<dim3(BB), 512, 0, stream>>>(covc, invLT);
  tmat_kernel <<<dim3(NN / 128, NN / 64, BB), 256, 0, stream>>>(covs, invLT, Tm);
  bias_kernel <<<dim3(BB), 512, 0, stream>>>(Tm, mc, ms, bias);
  final_kernel<<<dim3(MM / 128, NN / 64, BB), 256, 0, stream>>>(Tm, cont, bias, out);
}